// StartPointModel2_90074054131988
// MI455X (gfx1250) — compile-verified
//
#include <hip/hip_runtime.h>
#include <math.h>

// ---------------------------------------------------------------------------
// Problem dims (fixed by reference)
// ---------------------------------------------------------------------------
#define H_   256
#define B_   32
#define T_   512
#define F_   1024
#define G_   1024          // 4*H gate width
#define STEPS_ 64          // decoder steps
#define V_   3

typedef float v2f __attribute__((ext_vector_type(2)));
typedef float v8f __attribute__((ext_vector_type(8)));
typedef unsigned int u32x4 __attribute__((ext_vector_type(4)));
typedef int i32x4 __attribute__((ext_vector_type(4)));
typedef int i32x8 __attribute__((ext_vector_type(8)));

#if defined(__has_builtin)
#if __has_builtin(__builtin_amdgcn_tensor_load_to_lds) && \
    __has_builtin(__builtin_amdgcn_s_wait_tensorcnt)
#define HAVE_TDM 1
#endif
#endif
#ifndef HAVE_TDM
#define HAVE_TDM 0
#endif

static __device__ __forceinline__ v8f wmma4(v2f a, v2f b, v8f c) {
  // D = A(16x4 f32) * B(4x16 f32) + C(16x16 f32)
  return __builtin_amdgcn_wmma_f32_16x16x4_f32(
      /*neg_a=*/false, a, /*neg_b=*/false, b,
      /*c_mod=*/(short)0, c, /*reuse_a=*/false, /*reuse_b=*/false);
}

static __device__ __forceinline__ float sigm(float x) {
  return 1.0f / (1.0f + __expf(-x));
}

#if HAVE_TDM
// ---------------------------------------------------------------------------
// Issue one TDM 2D tile load: d0 contiguous elements (f32) x d1 rows with a
// row stride of `stride_elems`, into LDS at byte offset `lds_off`, inserting
// 1 DWORD of LDS padding every 32 DWORDs stored (pad_interval code 4,
// pad_amount code 0) -> produces a [d1][d0+1]-float padded LDS tile that is
// bank-conflict-free for WMMA operand reads.  D# packing per CDNA5 ISA
// "Tensor DMA Descriptor" groups 0/1; groups 2/3 unused (2D tensor).
// All arguments must be wave-uniform (descriptor lives in SGPRs).
// This toolchain exposes the 6-arg builtin form:
//   (u32x4 g0, i32x8 g1, i32x4 g2, i32x4 g3, i32x8 gX, i32 cpol)
// ---------------------------------------------------------------------------
static __device__ __forceinline__ void tdm_load_tile_2d(
    const float* g, unsigned int lds_off, int d0, int d1, long stride_elems)
{
  unsigned long long ga = (unsigned long long)(uintptr_t)g;
  u32x4 g0;
  g0[0] = 1u;                                   // count=1 valid user descriptor
  g0[1] = lds_off;                              // lds_addr (bytes)
  g0[2] = (unsigned int)ga;                     // global_addr[31:0]
  g0[3] = (unsigned int)((ga >> 32) & 0x01FFFFFFu) | (2u << 30); // [56:32]|type=2

  unsigned int td0 = (unsigned int)d0;          // tensor_dim0 == tile_dim0
  unsigned int td1 = (unsigned int)d1;          // tensor_dim1 == tile_dim1
  unsigned long long st0 = (unsigned long long)stride_elems; // data_size units
  i32x8 g1;
  g1[0] = (int)((2u << 16)      // data_size = 2 -> 4-byte elements
              | (1u << 20)      // pad_enable
              | (4u << 22)      // pad_interval code 4 -> every 32 DWORDs
              | (0u << 25));    // pad_amount code 0 -> 1 DWORD
  g1[1] = (int)((td0 & 0xFFFFu) << 16);                          // tdim0[15:0]
  g1[2] = (int)(((td0 >> 16) & 0xFFFFu) | ((td1 & 0xFFFFu) << 16));
  g1[3] = (int)(((td1 >> 16) & 0xFFFFu) | (((unsigned)d0 & 0xFFFFu) << 16)); // tile_dim0
  g1[4] = (int)((unsigned)d1 & 0xFFFFu);                         // tile_dim1, tile_dim2=0
  g1[5] = (int)(unsigned int)(st0 & 0xFFFFFFFFu);                // stride0[31:0]
  g1[6] = (int)(unsigned int)((st0 >> 32) & 0xFFFFu);            // stride0[47:32]
  g1[7] = 0;

  i32x4 z4 = {0, 0, 0, 0};
  i32x8 z8 = {0, 0, 0, 0, 0, 0, 0, 0};
  __builtin_amdgcn_tensor_load_to_lds(g0, g1, z4, z4, z8, 0);
}
#endif

// ---------------------------------------------------------------------------
// Kernel A: Gx[t*B+b, n] = cnn_feats[t,b,:] . Wih[n,:] + bias[n]
// M=16384, N=1024, K=1024.  Block = 256 thr (8 waves), tile 64(M) x 128(N).
// Each wave: 2x2 grid of 16x16 WMMA tiles (32x32).  Tiles staged to LDS by
// the Tensor Data Mover (wave 0 issues, all waves consume after barrier).
// ---------------------------------------------------------------------------
__global__ __launch_bounds__(256) void gates_gemm(
    const float* __restrict__ A,
    const float* __restrict__ W0, const float* __restrict__ bias0,
    const float* __restrict__ W1, const float* __restrict__ bias1,
    float* __restrict__ out0, float* __restrict__ out1, int dir_base)
{
  const int dir = dir_base + blockIdx.z;
  const float* __restrict__ W    = dir ? W1 : W0;
  const float* __restrict__ bias = dir ? bias1 : bias0;
  float* __restrict__ out        = dir ? out1 : out0;

  __shared__ float As[64][33];    // [m][k],  row stride 33 (TDM-padded)
  __shared__ float Bs[128][33];   // [n][k],  row stride 33 (TDM-padded)

  const int tid  = threadIdx.x;
  const int lane = tid & 31;
  const int w    = tid >> 5;
  const int wm   = w & 1;                 // wave row (0..1)
  const int wn   = w >> 1;                // wave col (0..3)
  const int mBase = blockIdx.y * 64;
  const int nBase = blockIdx.x * 128;
  const int row  = lane & 15;             // M row / N col within a 16-tile
  const int ksel = (lane >> 4) << 1;      // lanes 16-31 hold K+2,K+3
  const int hi8  = (lane >> 4) << 3;      // C store: VGPR r -> rows r, r+8

  v8f acc00 = {}, acc01 = {}, acc10 = {}, acc11 = {};

  for (int kb = 0; kb < F_; kb += 32) {
#if HAVE_TDM
    if (w == 0) {   // one wave drives the DMA engine; EXEC-independent op
      tdm_load_tile_2d(A + (size_t)mBase * F_ + kb,
                       (unsigned int)(uintptr_t)&As[0][0], 32, 64, F_);
      tdm_load_tile_2d(W + (size_t)nBase * F_ + kb,
                       (unsigned int)(uintptr_t)&Bs[0][0], 32, 128, F_);
      __builtin_amdgcn_s_wait_tensorcnt(0);
    }
#else
    for (int i = tid; i < 64 * 32; i += 256) {
      int m = i >> 5, k = i & 31;
      As[m][k] = A[(size_t)(mBase + m) * F_ + kb + k];
    }
    for (int i = tid; i < 128 * 32; i += 256) {
      int n = i >> 5, k = i & 31;
      Bs[n][k] = W[(size_t)(nBase + n) * F_ + kb + k];   // B[k][n] = Wih[n][k]
    }
#endif
    __syncthreads();
#pragma unroll
    for (int k4 = 0; k4 < 8; ++k4) {
      int kk = (k4 << 2) + ksel;
      v2f a0 = { As[wm * 32 + row][kk],      As[wm * 32 + row][kk + 1] };
      v2f a1 = { As[wm * 32 + 16 + row][kk], As[wm * 32 + 16 + row][kk + 1] };
      v2f b0 = { Bs[wn * 32 + row][kk],      Bs[wn * 32 + row][kk + 1] };
      v2f b1 = { Bs[wn * 32 + 16 + row][kk], Bs[wn * 32 + 16 + row][kk + 1] };
      acc00 = wmma4(a0, b0, acc00);
      acc01 = wmma4(a0, b1, acc01);
      acc10 = wmma4(a1, b0, acc10);
      acc11 = wmma4(a1, b1, acc11);
    }
    __syncthreads();
  }

#pragma unroll
  for (int r = 0; r < 8; ++r) {
    int m0 = mBase + wm * 32 + hi8 + r;
    int m1 = m0 + 16;
    int n0 = nBase + wn * 32 + row;
    int n1 = n0 + 16;
    out[(size_t)m0 * G_ + n0] = acc00[r] + bias[n0];
    out[(size_t)m0 * G_ + n1] = acc01[r] + bias[n1];
    out[(size_t)m1 * G_ + n0] = acc10[r] + bias[n0];
    out[(size_t)m1 * G_ + n1] = acc11[r] + bias[n1];
  }
}

// ---------------------------------------------------------------------------
// Kernel B: encoder recurrence. One 1024-thread (32-wave) workgroup per
// direction. h,c and the 32x1024 gate staging buffer live in LDS (197 KB,
// possible only on CDNA5's 320 KB WGP LDS). Whh streamed from L2 each step.
// Wave w owns gate columns [32w, 32w+32): 2x2 WMMA tiles over M=32.
// ---------------------------------------------------------------------------
__global__ __launch_bounds__(1024) void enc_rec(
    const float* __restrict__ gx0, const float* __restrict__ gx1,
    const float* __restrict__ Whh0, const float* __restrict__ Whh1,
    float* __restrict__ hf, float* __restrict__ cf,
    float* __restrict__ hb, float* __restrict__ cb, int dir_base)
{
  const int dir = dir_base + blockIdx.x;
  const float* __restrict__ gx  = dir ? gx1 : gx0;
  const float* __restrict__ Whh = dir ? Whh1 : Whh0;
  float* __restrict__ hout = dir ? hb : hf;
  float* __restrict__ cout = dir ? cb : cf;

  __shared__ float hS[32][257];     // h[b][j], stride 257 -> conflict-free
  __shared__ float cS[32][256];
  __shared__ float gS[32][1025];    // gates staging

  const int tid  = threadIdx.x;
  const int lane = tid & 31;
  const int w    = tid >> 5;              // 0..31
  const int row  = lane & 15;
  const int ksel = (lane >> 4) << 1;
  const int hi8  = (lane >> 4) << 3;
  const int n0   = w * 32 + row;          // gate column of first N-tile
  const int n1   = n0 + 16;

  for (int e = tid; e < 32 * 256; e += 1024) {
    hS[e >> 8][e & 255] = 0.0f;
    cS[e >> 8][e & 255] = 0.0f;
  }
  __syncthreads();

  for (int s = 0; s < T_; ++s) {
    const int t = dir ? (T_ - 1 - s) : s;
    const float* __restrict__ gxt = gx + (size_t)t * B_ * G_;

    // prefetch next step's precomputed gates (128 KB = 1024 cachelines)
    if (s + 1 < T_) {
      const int tn = dir ? (T_ - 2 - s) : (s + 1);
      __builtin_prefetch(gx + (size_t)tn * B_ * G_ + (size_t)tid * 32, 0, 1);
    }

    v8f acc00 = {}, acc01 = {}, acc10 = {}, acc11 = {};
#pragma unroll 8
    for (int k = 0; k < H_; k += 4) {
      int kk = k + ksel;
      v2f a0 = { hS[row][kk],      hS[row][kk + 1] };
      v2f a1 = { hS[16 + row][kk], hS[16 + row][kk + 1] };
      v2f b0 = *(const v2f*)(Whh + (size_t)n0 * H_ + kk);   // B[k][n]=Whh[n][k]
      v2f b1 = *(const v2f*)(Whh + (size_t)n1 * H_ + kk);
      acc00 = wmma4(a0, b0, acc00);
      acc01 = wmma4(a0, b1, acc01);
      acc10 = wmma4(a1, b0, acc10);
      acc11 = wmma4(a1, b1, acc11);
    }

#pragma unroll
    for (int r = 0; r < 8; ++r) {
      int m0 = hi8 + r, m1 = m0 + 16;
      gS[m0][n0] = acc00[r] + gxt[m0 * G_ + n0];
      gS[m0][n1] = acc01[r] + gxt[m0 * G_ + n1];
      gS[m1][n0] = acc10[r] + gxt[m1 * G_ + n0];
      gS[m1][n1] = acc11[r] + gxt[m1 * G_ + n1];
    }
    __syncthreads();

    // elementwise LSTM cell update (PyTorch gate order i,f,g,o)
    for (int e = tid; e < 32 * 256; e += 1024) {
      int b = e >> 8, j = e & 255;
      float gi = gS[b][j];
      float gf = gS[b][256 + j];
      float gg = gS[b][512 + j];
      float go = gS[b][768 + j];
      float cn = sigm(gf) * cS[b][j] + sigm(gi) * tanhf(gg);
      cS[b][j] = cn;
      hS[b][j] = sigm(go) * tanhf(cn);
    }
    __syncthreads();
  }

  for (int e = tid; e < 32 * 256; e += 1024) {
    hout[e] = hS[e >> 8][e & 255];
    cout[e] = cS[e >> 8][e & 255];
  }
}

// ---------------------------------------------------------------------------
// Kernel C: 2-layer decoder, 64 steps, single 1024-thread workgroup.
// x, h0, c0, h1, c1 + gate buffer all in LDS (295 KB). Per step, each cell:
// gates = x@Wih.T + h@Whh.T + b via WMMA, then elementwise; finally the V=3
// projection and x <- h1 feedback.
// ---------------------------------------------------------------------------
__global__ __launch_bounds__(1024) void decoder(
    const float* __restrict__ hf, const float* __restrict__ cf,
    const float* __restrict__ hb, const float* __restrict__ cb,
    const float* __restrict__ Wih0, const float* __restrict__ Whh0,
    const float* __restrict__ b0v,
    const float* __restrict__ Wih1, const float* __restrict__ Whh1,
    const float* __restrict__ b1v,
    const float* __restrict__ linW, const float* __restrict__ linb,
    float* __restrict__ out)
{
  __shared__ float xS[32][257];
  __shared__ float h0S[32][257];
  __shared__ float h1S[32][257];
  __shared__ float c0S[32][256];
  __shared__ float c1S[32][256];
  __shared__ float gS[32][1025];

  const int tid  = threadIdx.x;
  const int lane = tid & 31;
  const int w    = tid >> 5;
  const int row  = lane & 15;
  const int ksel = (lane >> 4) << 1;
  const int hi8  = (lane >> 4) << 3;
  const int n0   = w * 32 + row;
  const int n1   = n0 + 16;

  for (int e = tid; e < 32 * 256; e += 1024) {
    int b = e >> 8, j = e & 255;
    xS[b][j]  = 0.0f;
    h0S[b][j] = hf[e];  c0S[b][j] = cf[e];    // layer0 init = fwd state
    h1S[b][j] = hb[e];  c1S[b][j] = cb[e];    // layer1 init = bwd state
  }
  __syncthreads();

  for (int s = 0; s < STEPS_; ++s) {
    // ---- layer 0: gates = x@Wih0.T + h0@Whh0.T + b0 ----
    {
      v8f acc00 = {}, acc01 = {}, acc10 = {}, acc11 = {};
#pragma unroll 4
      for (int k = 0; k < H_; k += 4) {
        int kk = k + ksel;
        v2f ax0 = { xS[row][kk],       xS[row][kk + 1] };
        v2f ax1 = { xS[16 + row][kk],  xS[16 + row][kk + 1] };
        v2f ah0 = { h0S[row][kk],      h0S[row][kk + 1] };
        v2f ah1 = { h0S[16 + row][kk], h0S[16 + row][kk + 1] };
        v2f bi0 = *(const v2f*)(Wih0 + (size_t)n0 * H_ + kk);
        v2f bi1 = *(const v2f*)(Wih0 + (size_t)n1 * H_ + kk);
        v2f bh0 = *(const v2f*)(Whh0 + (size_t)n0 * H_ + kk);
        v2f bh1 = *(const v2f*)(Whh0 + (size_t)n1 * H_ + kk);
        acc00 = wmma4(ax0, bi0, acc00);  acc00 = wmma4(ah0, bh0, acc00);
        acc01 = wmma4(ax0, bi1, acc01);  acc01 = wmma4(ah0, bh1, acc01);
        acc10 = wmma4(ax1, bi0, acc10);  acc10 = wmma4(ah1, bh0, acc10);
        acc11 = wmma4(ax1, bi1, acc11);  acc11 = wmma4(ah1, bh1, acc11);
      }
#pragma unroll
      for (int r = 0; r < 8; ++r) {
        int m0 = hi8 + r, m1 = m0 + 16;
        gS[m0][n0] = acc00[r] + b0v[n0];
        gS[m0][n1] = acc01[r] + b0v[n1];
        gS[m1][n0] = acc10[r] + b0v[n0];
        gS[m1][n1] = acc11[r] + b0v[n1];
      }
      __syncthreads();
      for (int e = tid; e < 32 * 256; e += 1024) {
        int b = e >> 8, j = e & 255;
        float gi = gS[b][j], gf = gS[b][256 + j];
        float gg = gS[b][512 + j], go = gS[b][768 + j];
        float cn = sigm(gf) * c0S[b][j] + sigm(gi) * tanhf(gg);
        c0S[b][j] = cn;
        h0S[b][j] = sigm(go) * tanhf(cn);
      }
      __syncthreads();
    }

    // ---- layer 1: gates = h0@Wih1.T + h1@Whh1.T + b1 ----
    {
      v8f acc00 = {}, acc01 = {}, acc10 = {}, acc11 = {};
#pragma unroll 4
      for (int k = 0; k < H_; k += 4) {
        int kk = k + ksel;
        v2f ax0 = { h0S[row][kk],      h0S[row][kk + 1] };
        v2f ax1 = { h0S[16 + row][kk], h0S[16 + row][kk + 1] };
        v2f ah0 = { h1S[row][kk],      h1S[row][kk + 1] };
        v2f ah1 = { h1S[16 + row][kk], h1S[16 + row][kk + 1] };
        v2f bi0 = *(const v2f*)(Wih1 + (size_t)n0 * H_ + kk);
        v2f bi1 = *(const v2f*)(Wih1 + (size_t)n1 * H_ + kk);
        v2f bh0 = *(const v2f*)(Whh1 + (size_t)n0 * H_ + kk);
        v2f bh1 = *(const v2f*)(Whh1 + (size_t)n1 * H_ + kk);
        acc00 = wmma4(ax0, bi0, acc00);  acc00 = wmma4(ah0, bh0, acc00);
        acc01 = wmma4(ax0, bi1, acc01);  acc01 = wmma4(ah0, bh1, acc01);
        acc10 = wmma4(ax1, bi0, acc10);  acc10 = wmma4(ah1, bh0, acc10);
        acc11 = wmma4(ax1, bi1, acc11);  acc11 = wmma4(ah1, bh1, acc11);
      }
#pragma unroll
      for (int r = 0; r < 8; ++r) {
        int m0 = hi8 + r, m1 = m0 + 16;
        gS[m0][n0] = acc00[r] + b1v[n0];
        gS[m0][n1] = acc01[r] + b1v[n1];
        gS[m1][n0] = acc10[r] + b1v[n0];
        gS[m1][n1] = acc11[r] + b1v[n1];
      }
      __syncthreads();
      for (int e = tid; e < 32 * 256; e += 1024) {
        int b = e >> 8, j = e & 255;
        float gi = gS[b][j], gf = gS[b][256 + j];
        float gg = gS[b][512 + j], go = gS[b][768 + j];
        float cn = sigm(gf) * c1S[b][j] + sigm(gi) * tanhf(gg);
        c1S[b][j] = cn;
        h1S[b][j] = sigm(go) * tanhf(cn);
      }
      __syncthreads();
    }

    // ---- projection y = h1 @ linW.T + linb, and x <- h1 feedback ----
    if (tid < B_ * V_) {
      int b = tid / V_, v = tid - b * V_;
      float acc = linb[v];
      for (int j = 0; j < H_; ++j) acc += h1S[b][j] * linW[v * H_ + j];
      out[(size_t)s * (B_ * V_) + tid] = acc;
    }
    for (int e = tid; e < 32 * 256; e += 1024) {
      xS[e >> 8][e & 255] = h1S[e >> 8][e & 255];
    }
    __syncthreads();
  }
}

// ---------------------------------------------------------------------------
// Host launcher
// ---------------------------------------------------------------------------
extern "C" void kernel_launch(void* const* d_in, const int* in_sizes, int n_in,
                              void* d_out, int out_size, void* d_ws, size_t ws_size,
                              hipStream_t stream) {
  (void)in_sizes; (void)n_in; (void)out_size;
  const float* feats = (const float*)d_in[0];
  const float* WihF  = (const float*)d_in[1];
  const float* WhhF  = (const float*)d_in[2];
  const float* bF    = (const float*)d_in[3];
  const float* WihB  = (const float*)d_in[4];
  const float* WhhB  = (const float*)d_in[5];
  const float* bB    = (const float*)d_in[6];
  const float* dWih0 = (const float*)d_in[7];
  const float* dWhh0 = (const float*)d_in[8];
  const float* db0   = (const float*)d_in[9];
  const float* dWih1 = (const float*)d_in[10];
  const float* dWhh1 = (const float*)d_in[11];
  const float* db1   = (const float*)d_in[12];
  const float* linW  = (const float*)d_in[13];
  const float* linb  = (const float*)d_in[14];
  float* out = (float*)d_out;
  float* ws  = (float*)d_ws;

  const size_t GX    = (size_t)T_ * B_ * G_;   // 16.78M floats per direction
  const size_t STATE = (size_t)4 * B_ * H_;    // hf,cf,hb,cb
  const bool both = ws_size >= (2 * GX + STATE) * sizeof(float);

  float* gx0 = ws;
  float* gx1 = both ? ws + GX : ws;            // alias in fallback mode
  float* st  = both ? ws + 2 * GX : ws + GX;
  float* hf = st, *cf = st + B_ * H_, *hb = st + 2 * B_ * H_, *cb = st + 3 * B_ * H_;

  const dim3 blkA(256), blkR(1024);
  if (both) {
    // both directions in parallel: full-GPU GEMM, then 2 concurrent
    // persistent-workgroup recurrences
    gates_gemm<<<dim3(G_ / 128, (T_ * B_) / 64, 2), blkA, 0, stream>>>(
        feats, WihF, bF, WihB, bB, gx0, gx1, 0);
    enc_rec<<<dim3(2), blkR, 0, stream>>>(gx0, gx1, WhhF, WhhB, hf, cf, hb, cb, 0);
  } else {
    // small-workspace fallback: process directions sequentially, one buffer
    gates_gemm<<<dim3(G_ / 128, (T_ * B_) / 64, 1), blkA, 0, stream>>>(
        feats, WihF, bF, WihB, bB, gx0, gx0, 0);
    enc_rec<<<dim3(1), blkR, 0, stream>>>(gx0, gx0, WhhF, WhhB, hf, cf, hb, cb, 0);
    gates_gemm<<<dim3(G_ / 128, (T_ * B_) / 64, 1), blkA, 0, stream>>>(
        feats, WihF, bF, WihB, bB, gx0, gx0, 1);
    enc_rec<<<dim3(1), blkR, 0, stream>>>(gx0, gx0, WhhF, WhhB, hf, cf, hb, cb, 1);
  }
  decoder<<<dim3(1), blkR, 0, stream>>>(hf, cf, hb, cb,
                                        dWih0, dWhh0, db0, dWih1, dWhh1, db1,
                                        linW, linb, out);
}